// MsaHmmLayer_6820408066849
// MI455X (gfx1250) — compile-verified
//
#include <hip/hip_runtime.h>
#include <math.h>

// HMM forward/backward (scaled, probability-space) for MI455X / gfx1250.
// NUM_MODEL=4, B=32, L=1024, S=26, Q=128.
//
// d_out = posterior log-probs (4*32*1024*128 floats) ++ loglik (4*32 floats).
// Workspace (floats): E (m,b,L,q) emissions, A=softmax(trans) rows,
// Bmat=softmax(emission) padded to 28 cols, pi=softmax(init). ~67.5 MB.
//
// Scan kernels: transition WMMA B-fragments preloaded in registers for all
// 1024 steps; alpha/beta tiles in LDS; emissions (and alpha, in backward)
// double-buffered via GLOBAL_LOAD_ASYNC_TO_LDS_B128 one step ahead; row
// normalization deferred one step and folded into the emission multiply;
// row sums via in-register shuffle reductions (2 barriers/step forward).

#define NM 4
#define BB 32
#define LL 1024
#define SS 26
#define QQ 128
#define SP 28           // S padded to multiple of 4 for WMMA K
#define TS 130          // LDS stride for matmul operand tiles (bank padding)
#define ETS 132         // LDS stride for async-staged tiles (16B-aligned rows)

typedef float v2f __attribute__((ext_vector_type(2)));
typedef float v8f __attribute__((ext_vector_type(8)));
typedef int   v4i __attribute__((ext_vector_type(4)));

#define E_SIZE  ((size_t)NM*BB*LL*QQ)
#define A_SIZE  ((size_t)NM*QQ*QQ)
#define B_SIZE  ((size_t)NM*QQ*SP)
#define POST_SIZE ((size_t)NM*BB*LL*QQ)

static __device__ __forceinline__ v8f wmma4(v2f a, v2f b, v8f c) {
  // V_WMMA_F32_16X16X4_F32 : D(16x16 f32) = A(16x4) * B(4x16) + C
  return __builtin_amdgcn_wmma_f32_16x16x4_f32(
      false, a, false, b, (short)0, c, false, false);
}

static __device__ __forceinline__ float wred_max(float v) {
  #pragma unroll
  for (int o = 16; o > 0; o >>= 1) v = fmaxf(v, __shfl_xor(v, o, 32));
  return v;
}
static __device__ __forceinline__ float wred_sum(float v) {
  #pragma unroll
  for (int o = 16; o > 0; o >>= 1) v += __shfl_xor(v, o, 32);
  return v;
}

// One 128-bit async Global->LDS copy (ASYNCcnt-tracked).
static __device__ __forceinline__ void async_b128(const float* g, float* l) {
  v4i* gg = (v4i*)g;   // drop const, reinterpret (generic AS)
  v4i* ll = (v4i*)l;
  __builtin_amdgcn_global_load_async_to_lds_b128(
      (__attribute__((address_space(1))) v4i*)gg,
      (__attribute__((address_space(3))) v4i*)ll,
      0, 0);
}

// Async-stage one 16-row x 16-col tile slice (this wave's columns) from a
// (row-stride LL*QQ) global tile into LDS (row-stride ETS). 2x B128 per lane.
static __device__ __forceinline__ void stage_tile_async(
    const float* gbase /* &X[(rowbase)*LL + t]*QQ */, float* lbase,
    int lane, int nt) {
  #pragma unroll
  for (int i = 0; i < 2; ++i) {
    const int id = lane + 32*i;        // 0..63
    const int r  = id >> 2;            // 0..15 tile row
    const int c  = nt*16 + (id & 3)*4; // 16B-aligned column group
    async_b128(gbase + (size_t)r*LL*QQ + c, lbase + r*ETS + c);
  }
}
static __device__ __forceinline__ void wait_async0() {
  asm volatile("s_wait_asynccnt 0x0" ::: "memory");
}
static __device__ __forceinline__ void wait_async2() {
  asm volatile("s_wait_asynccnt 0x2" ::: "memory");
}
static __device__ __forceinline__ void wait_async4() {
  asm volatile("s_wait_asynccnt 0x4" ::: "memory");
}

// ---------------------------------------------------------------------------
// 1) Row softmaxes: A, Bmat (padded to 28), pi. One wave per row.
// ---------------------------------------------------------------------------
__global__ __launch_bounds__(32) void prep_kernel(
    const float* __restrict__ trans, const float* __restrict__ emis,
    const float* __restrict__ initl,
    float* __restrict__ Aw, float* __restrict__ Bw, float* __restrict__ Pw) {
  const int b = blockIdx.x, lane = threadIdx.x;
  if (b < NM*QQ || b >= 2*NM*QQ) {                     // 128-wide softmax
    const bool isA = (b < NM*QQ);
    const int r = isA ? b : (b - 2*NM*QQ);
    const float* row = (isA ? trans : initl) + (size_t)r*QQ;
    float v0 = row[lane], v1 = row[lane+32], v2 = row[lane+64], v3 = row[lane+96];
    float mx = wred_max(fmaxf(fmaxf(v0, v1), fmaxf(v2, v3)));
    float e0 = expf(v0-mx), e1 = expf(v1-mx), e2 = expf(v2-mx), e3 = expf(v3-mx);
    float s = wred_sum(e0+e1+e2+e3);
    float* o = (isA ? Aw : Pw) + (size_t)r*QQ;
    o[lane] = e0/s; o[lane+32] = e1/s; o[lane+64] = e2/s; o[lane+96] = e3/s;
  } else {                                             // 26-wide softmax, pad
    const int r = b - NM*QQ;
    const float* row = emis + (size_t)r*SS;
    float v = (lane < SS) ? row[lane] : -1e30f;
    float mx = wred_max(v);
    float e = (lane < SS) ? expf(v-mx) : 0.0f;
    float s = wred_sum(e);
    float* o = Bw + (size_t)r*SP;
    if (lane < SS) o[lane] = e/s;
    else if (lane < SP) o[lane] = 0.0f;
  }
}

// ---------------------------------------------------------------------------
// 2) Emission GEMM: E[m,row,q] = sum_s inputs[m,row,s] * Bmat[m,q,s]
//    M=16 per block, N=128 (8 waves), K=26 padded to 28 (7 WMMA).
// ---------------------------------------------------------------------------
__global__ __launch_bounds__(256) void egemm_kernel(
    const float* __restrict__ inputs, const float* __restrict__ Bw,
    float* __restrict__ Ew) {
  __shared__ float IN[16*SP];
  __shared__ float BT[SP*TS];
  const int m = blockIdx.x, rt = blockIdx.y;
  const int tid = threadIdx.x, lane = tid & 31;
  const int nt = tid >> 5, hi = lane >> 4, l15 = lane & 15;

  const float* gin = inputs + ((size_t)m*BB*LL + (size_t)rt*16)*SS;
  for (int i = tid; i < 16*SS; i += 256) IN[(i/SS)*SP + (i%SS)] = gin[i];
  if (tid < 32) IN[(tid>>1)*SP + SS + (tid&1)] = 0.0f;
  for (int i = tid; i < SP*QQ; i += 256) {
    const int s = i >> 7, q = i & 127;
    BT[s*TS + q] = Bw[((size_t)m*QQ + q)*SP + s];
  }
  __syncthreads();

  v8f acc = {0.f,0.f,0.f,0.f,0.f,0.f,0.f,0.f};
  #pragma unroll
  for (int kc = 0; kc < 7; ++kc) {
    const int k = 4*kc + 2*hi;
    v2f a, bb;
    a.x  = IN[l15*SP + k];            a.y  = IN[l15*SP + k + 1];
    bb.x = BT[k*TS + nt*16 + l15];    bb.y = BT[(k+1)*TS + nt*16 + l15];
    acc = wmma4(a, bb, acc);
  }
  float* Eo = Ew + ((size_t)m*BB*LL + (size_t)rt*16)*QQ;
  #pragma unroll
  for (int j = 0; j < 8; ++j)
    Eo[(size_t)(j + 8*hi)*QQ + nt*16 + l15] = acc[j];
}

// ---------------------------------------------------------------------------
// 3) Forward scan: one block per (model, batch-half), 8 waves, 16 rows.
//    raw_t = (raw_{t-1} x A) .* E_t .* rinv_{t-1}[row]   (deferred norm)
//    out <- raw_t * rinv_t ; loglik += log(rowsum_t).
// ---------------------------------------------------------------------------
__global__ __launch_bounds__(256) void hmm_forward(
    const float* __restrict__ Ew, const float* __restrict__ Aw,
    const float* __restrict__ Pw, float* __restrict__ out) {
  __shared__ __align__(16) float abuf[2][16*TS];   // raw alpha tiles
  __shared__ __align__(16) float ebuf[2][16*ETS];  // async-staged emissions
  __shared__ float part[16][8];
  __shared__ float rinv[2][16];
  __shared__ float lacc[16];

  const int m = blockIdx.x >> 1, bh = blockIdx.x & 1;
  const int tid = threadIdx.x, lane = tid & 31;
  const int nt = tid >> 5, hi = lane >> 4, l15 = lane & 15;
  const int c0 = nt*16 + l15;                      // this lane's column
  const size_t rowbase = (size_t)m*BB + bh*16;
  const float* aw = Aw + (size_t)m*QQ*QQ;

  // Transition B-fragments: constant across all 1024 steps.
  v2f bf[32];
  #pragma unroll
  for (int kc = 0; kc < 32; ++kc) {
    const int K = 4*kc + 2*hi, N = nt*16 + l15;
    bf[kc].x = aw[(size_t)K*QQ + N];
    bf[kc].y = aw[(size_t)(K+1)*QQ + N];
  }
  if (tid < 16) lacc[tid] = 0.0f;

  // Kick off async staging of E_1 while we do t=0 directly.
  stage_tile_async(Ew + (rowbase*LL + 1)*QQ, ebuf[1], lane, nt);

  float val[8];
  #pragma unroll
  for (int j = 0; j < 8; ++j) {                    // raw alpha0 = pi .* E0
    const int M = j + 8*hi;
    val[j] = Pw[m*QQ + c0] * Ew[((rowbase + M)*LL + 0)*QQ + c0];
    abuf[0][M*TS + c0] = val[j];
  }

  for (int t = 0; t < LL; ++t) {
    const int cb = t & 1;
    if (t > 0) {
      const int pb = (t-1) & 1;
      if (t + 1 < LL)                              // prefetch E_{t+1}
        stage_tile_async(Ew + (rowbase*LL + t + 1)*QQ, ebuf[(t+1)&1], lane, nt);
      const float* cur = abuf[pb];
      v8f acc0 = {0.f,0.f,0.f,0.f,0.f,0.f,0.f,0.f};
      v8f acc1 = acc0;
      #pragma unroll
      for (int kc = 0; kc < 32; kc += 2) {         // 2 chains hide XDL latency
        v2f a0, a1;
        const int k0 = 4*kc + 2*hi, k1 = 4*(kc+1) + 2*hi;
        a0.x = cur[l15*TS + k0]; a0.y = cur[l15*TS + k0 + 1];
        a1.x = cur[l15*TS + k1]; a1.y = cur[l15*TS + k1 + 1];
        acc0 = wmma4(a0, bf[kc],   acc0);
        acc1 = wmma4(a1, bf[kc+1], acc1);
      }
      if (t + 1 < LL) wait_async2(); else wait_async0();  // E_t tile ready
      #pragma unroll
      for (int j = 0; j < 8; ++j) {
        const int M = j + 8*hi;
        val[j] = (acc0[j] + acc1[j]) * rinv[pb][M] * ebuf[cb][M*ETS + c0];
        abuf[cb][M*TS + c0] = val[j];
      }
    }
    // shuffle row-sum partials within each 16-lane half (rows j+8*hi)
    #pragma unroll
    for (int j = 0; j < 8; ++j) {
      float p = val[j];
      p += __shfl_xor(p, 1, 32); p += __shfl_xor(p, 2, 32);
      p += __shfl_xor(p, 4, 32); p += __shfl_xor(p, 8, 32);
      if (l15 == 0) part[j + 8*hi][nt] = p;
    }
    __syncthreads();
    if (tid < 16) {
      float s = 0.0f;
      #pragma unroll
      for (int ch = 0; ch < 8; ++ch) s += part[tid][ch];
      rinv[cb][tid] = 1.0f / s;
      lacc[tid] += logf(s);
    }
    __syncthreads();
    #pragma unroll
    for (int j = 0; j < 8; ++j) {                  // emit normalized alpha
      const int M = j + 8*hi;
      out[((rowbase + M)*LL + t)*QQ + c0] = val[j] * rinv[cb][M];
    }
  }
  __syncthreads();
  if (tid < 16) out[POST_SIZE + rowbase + tid] = lacc[tid];
}

// ---------------------------------------------------------------------------
// 4) Backward scan + posterior (overwrites alpha in d_out).
//    braw_{t-1} = ((E_t .* braw_t) .* rinvB_t[row]) x A^T
//    posterior_t = log( (alpha_t .* braw_t) / rowsum )   (self-normalizing)
// ---------------------------------------------------------------------------
__global__ __launch_bounds__(256) void hmm_backward(
    const float* __restrict__ Ew, const float* __restrict__ Aw,
    float* __restrict__ out) {
  __shared__ __align__(16) float bbuf[2][16*TS];   // raw beta tiles
  __shared__ __align__(16) float scr[16*TS];       // U tile (matmul A operand)
  __shared__ __align__(16) float stA[2][16*ETS];   // async-staged alpha
  __shared__ __align__(16) float stE[2][16*ETS];   // async-staged emissions
  __shared__ float part[16][8];
  __shared__ float rp[16];
  __shared__ float rinvB[2][16];

  const int m = blockIdx.x >> 1, bh = blockIdx.x & 1;
  const int tid = threadIdx.x, lane = tid & 31;
  const int nt = tid >> 5, hi = lane >> 4, l15 = lane & 15;
  const int c0 = nt*16 + l15;
  const size_t rowbase = (size_t)m*BB + bh*16;
  const float* aw = Aw + (size_t)m*QQ*QQ;

  // Transposed transition fragments: B[k][n] = A[n][k].
  v2f bf[32];
  #pragma unroll
  for (int kc = 0; kc < 32; ++kc) {
    const int K = 4*kc + 2*hi, N = nt*16 + l15;
    bf[kc].x = aw[(size_t)N*QQ + K];
    bf[kc].y = aw[(size_t)N*QQ + K + 1];
  }

  // Stage alpha/E for t = LL-1; init beta_raw = 1, rinvB = 1/Q.
  stage_tile_async(out + (rowbase*LL + (LL-1))*QQ, stA[(LL-1)&1], lane, nt);
  stage_tile_async(Ew  + (rowbase*LL + (LL-1))*QQ, stE[(LL-1)&1], lane, nt);
  int bi = 0;
  #pragma unroll
  for (int j = 0; j < 8; ++j) bbuf[0][(j + 8*hi)*TS + c0] = 1.0f;
  if (tid < 16) rinvB[0][tid] = 1.0f/(float)QQ;
  __syncthreads();

  for (int t = LL - 1; t >= 0; --t) {
    const int pa = t & 1;
    if (t > 0) {                                   // prefetch step t-1 tiles
      stage_tile_async(out + (rowbase*LL + (t-1))*QQ, stA[(t-1)&1], lane, nt);
      stage_tile_async(Ew  + (rowbase*LL + (t-1))*QQ, stE[(t-1)&1], lane, nt);
      wait_async4();                               // tiles for t are ready
    } else {
      wait_async0();
    }
    // posterior: prod = alpha_hat .* beta_raw (kept in registers)
    float prod[8];
    #pragma unroll
    for (int j = 0; j < 8; ++j) {
      const int M = j + 8*hi;
      prod[j] = stA[pa][M*ETS + c0] * bbuf[bi][M*TS + c0];
      float p = prod[j];
      p += __shfl_xor(p, 1, 32); p += __shfl_xor(p, 2, 32);
      p += __shfl_xor(p, 4, 32); p += __shfl_xor(p, 8, 32);
      if (l15 == 0) part[M][nt] = p;
    }
    __syncthreads();
    if (tid < 16) {
      float s = 0.0f;
      #pragma unroll
      for (int ch = 0; ch < 8; ++ch) s += part[tid][ch];
      rp[tid] = 1.0f / s;
    }
    __syncthreads();
    #pragma unroll
    for (int j = 0; j < 8; ++j) {
      const int M = j + 8*hi;
      out[((rowbase + M)*LL + t)*QQ + c0] =
          logf(fmaxf(prod[j] * rp[M], 1e-37f));
    }
    if (t == 0) break;

    // U = E_t .* beta_norm  ->  scr (read by all waves as matmul A operand)
    #pragma unroll
    for (int j = 0; j < 8; ++j) {
      const int M = j + 8*hi;
      scr[M*TS + c0] = stE[pa][M*ETS + c0] * bbuf[bi][M*TS + c0] * rinvB[bi][M];
    }
    __syncthreads();
    v8f acc0 = {0.f,0.f,0.f,0.f,0.f,0.f,0.f,0.f};
    v8f acc1 = acc0;
    #pragma unroll
    for (int kc = 0; kc < 32; kc += 2) {
      v2f a0, a1;
      const int k0 = 4*kc + 2*hi, k1 = 4*(kc+1) + 2*hi;
      a0.x = scr[l15*TS + k0]; a0.y = scr[l15*TS + k0 + 1];
      a1.x = scr[l15*TS + k1]; a1.y = scr[l15*TS + k1 + 1];
      acc0 = wmma4(a0, bf[kc],   acc0);
      acc1 = wmma4(a1, bf[kc+1], acc1);
    }
    const int nb = bi ^ 1;
    #pragma unroll
    for (int j = 0; j < 8; ++j) {
      const int M = j + 8*hi;
      const float v = acc0[j] + acc1[j];
      bbuf[nb][M*TS + c0] = v;
      float p = v;
      p += __shfl_xor(p, 1, 32); p += __shfl_xor(p, 2, 32);
      p += __shfl_xor(p, 4, 32); p += __shfl_xor(p, 8, 32);
      if (l15 == 0) part[M][nt] = p;
    }
    __syncthreads();
    if (tid < 16) {
      float s = 0.0f;
      #pragma unroll
      for (int ch = 0; ch < 8; ++ch) s += part[tid][ch];
      rinvB[nb][tid] = 1.0f / s;   // consumed after next step's barriers
    }
    bi = nb;
  }
}

// ---------------------------------------------------------------------------
extern "C" void kernel_launch(void* const* d_in, const int* in_sizes, int n_in,
                              void* d_out, int out_size, void* d_ws, size_t ws_size,
                              hipStream_t stream) {
  const float* inputs = (const float*)d_in[0];   // (4,32,1024,26)
  const float* trans  = (const float*)d_in[1];   // (4,128,128)
  const float* initl  = (const float*)d_in[2];   // (4,128)
  const float* emis   = (const float*)d_in[3];   // (4,128,26)
  float* out = (float*)d_out;
  float* ws  = (float*)d_ws;

  float* Ew = ws;
  float* Aw = Ew + E_SIZE;
  float* Bw = Aw + A_SIZE;
  float* Pw = Bw + B_SIZE;

  hipLaunchKernelGGL(prep_kernel, dim3(2*NM*QQ + NM), dim3(32), 0, stream,
                     trans, emis, initl, Aw, Bw, Pw);
  hipLaunchKernelGGL(egemm_kernel, dim3(NM, (BB*LL)/16), dim3(256), 0, stream,
                     inputs, Bw, Ew);
  hipLaunchKernelGGL(hmm_forward, dim3(2*NM), dim3(256), 0, stream,
                     Ew, Aw, Pw, out);
  hipLaunchKernelGGL(hmm_backward, dim3(2*NM), dim3(256), 0, stream,
                     Ew, Aw, out);
}